// SPGP_37099927502855
// MI455X (gfx1250) — compile-verified
//
#include <hip/hip_runtime.h>
#include <stdint.h>

typedef __attribute__((ext_vector_type(2))) float v2f;
typedef __attribute__((ext_vector_type(4))) float fvec4;
typedef __attribute__((ext_vector_type(8))) float v8f;

#define T_STEPS  12
#define TP       16      // padded t-stride in LDS (zeros in slots 12..15)
#define HW       3136    // 56*56
#define CHUNK    256     // pixels per LDS tile
#define NTHREADS 256     // 8 wave32

// ---------------- JAX threefry2x32 (exact) ----------------
__device__ __forceinline__ uint32_t rotl32(uint32_t x, uint32_t d) {
  return (x << d) | (x >> (32u - d));
}

struct U2 { uint32_t a, b; };

__device__ __forceinline__ U2 threefry2x32(uint32_t k0, uint32_t k1,
                                           uint32_t x0, uint32_t x1) {
  const uint32_t k2 = k0 ^ k1 ^ 0x1BD11BDAu;
  const uint32_t rA[4] = {13u, 15u, 26u, 6u};
  const uint32_t rB[4] = {17u, 29u, 16u, 24u};
  x0 += k0; x1 += k1;
#pragma unroll
  for (int j = 0; j < 4; ++j) { x0 += x1; x1 = rotl32(x1, rA[j]); x1 ^= x0; }
  x0 += k1; x1 += k2 + 1u;
#pragma unroll
  for (int j = 0; j < 4; ++j) { x0 += x1; x1 = rotl32(x1, rB[j]); x1 ^= x0; }
  x0 += k2; x1 += k0 + 2u;
#pragma unroll
  for (int j = 0; j < 4; ++j) { x0 += x1; x1 = rotl32(x1, rA[j]); x1 ^= x0; }
  x0 += k0; x1 += k1 + 3u;
#pragma unroll
  for (int j = 0; j < 4; ++j) { x0 += x1; x1 = rotl32(x1, rB[j]); x1 ^= x0; }
  x0 += k1; x1 += k2 + 4u;
#pragma unroll
  for (int j = 0; j < 4; ++j) { x0 += x1; x1 = rotl32(x1, rA[j]); x1 ^= x0; }
  x0 += k2; x1 += k0 + 5u;
  U2 r; r.a = x0; r.b = x1; return r;
}

// key_t from jax.random.split(key(1), 12): bits = threefry(root, iota(24))
__device__ __forceinline__ void jax_step_key(int t, uint32_t* k0, uint32_t* k1) {
  const uint32_t ut = (uint32_t)t;
  if (t < 6) {   // bits[2t], bits[2t+1] both come from out0 half
    *k0 = threefry2x32(0u, 1u, 2u * ut,      2u * ut + 12u).a;
    *k1 = threefry2x32(0u, 1u, 2u * ut + 1u, 2u * ut + 13u).a;
  } else {       // both come from out1 half
    *k0 = threefry2x32(0u, 1u, 2u * ut - 12u, 2u * ut).b;
    *k1 = threefry2x32(0u, 1u, 2u * ut - 11u, 2u * ut + 1u).b;
  }
}

// jax.random.uniform(key_t, (16,256,1,1)): bits over iota(4096) split in halves
__device__ __forceinline__ float jax_uniform4096(uint32_t k0, uint32_t k1, uint32_t i) {
  uint32_t bits;
  if (i < 2048u) bits = threefry2x32(k0, k1, i, i + 2048u).a;
  else           bits = threefry2x32(k0, k1, i - 2048u, i).b;
  union { uint32_t u; float f; } cvt;
  cvt.u = (bits >> 9) | 0x3f800000u;
  float r = cvt.f - 1.0f;
  return r < 0.0f ? 0.0f : r;   // lax.max(minval=0, ...)
}

// ---------------- fused GAP (via WMMA) + LIF scan ----------------
__global__ __launch_bounds__(NTHREADS)
void snn_gap_lif_kernel(const float* __restrict__ x,
                        const float* __restrict__ decay_p,
                        const float* __restrict__ vth_p,
                        float* __restrict__ out) {
  __shared__ float tile[CHUNK * TP];      // 16 KB: [pixel][t] padded to 16
  __shared__ float partial[8][TP];        // per-wave D column-0 dumps

  const int bc   = blockIdx.x;            // (b*256 + c), 4096 blocks
  const int tid  = threadIdx.x;
  const int wave = tid >> 5;
  const int lane = tid & 31;

  // zero pad slots t=12..15 once (loader never writes them)
  tile[tid * TP + 12] = 0.0f;
  tile[tid * TP + 13] = 0.0f;
  tile[tid * TP + 14] = 0.0f;
  tile[tid * TP + 15] = 0.0f;
  __syncthreads();

  // WMMA operands: B = ones (4x16), D accumulates per-row (=per-t) sums
  v2f bones; bones.x = 1.0f; bones.y = 1.0f;
  v8f acc = {};

  // contiguous 147KB slab for this (b,c): HW*T floats, float4-aligned
  const fvec4* xg = reinterpret_cast<const fvec4*>(x) +
                    (size_t)bc * (size_t)(HW * T_STEPS / 4);

  const int row   = lane & 15;            // A-matrix row (t index)
  const int khalf = (lane >> 4) * 2;      // K offset for this half-wave

  for (int chunk = 0; chunk < 13; ++chunk) {
    const int pix0 = chunk * CHUNK;
    int np = HW - pix0; if (np > CHUNK) np = CHUNK;   // 256 or 64
    const int nvec = np * 3;                          // np*12/4 float4s

    // streaming coalesced global -> LDS transpose into stride-16 layout
    for (int g = tid; g < nvec; g += NTHREADS) {
      fvec4 v = __builtin_nontemporal_load(&xg[(size_t)pix0 * 3 + g]);
      const int e0 = g * 4;
#pragma unroll
      for (int j = 0; j < 4; ++j) {
        const int e   = e0 + j;
        const int pix = e / T_STEPS;
        const int t   = e - pix * T_STEPS;
        tile[pix * TP + t] = v[j];
      }
    }
    __syncthreads();

    // A[t][k] = tile[(4g + k)*16 + t]; D[t][*] += sum_k A  (B = ones)
    const int ngroups = np >> 2;                      // wave-uniform
    for (int g = wave; g < ngroups; g += 8) {         // wave-uniform branch
      const float* base = &tile[(g * 4 + khalf) * TP + row];
      v2f a; a.x = base[0]; a.y = base[TP];
      acc = __builtin_amdgcn_wmma_f32_16x16x4_f32(
          false, a, false, bones, (short)0, acc, false, false);
    }
    __syncthreads();
  }

  // drain column 0: lane 0 holds rows 0..7, lane 16 holds rows 8..15
  if (lane == 0) {
#pragma unroll
    for (int r = 0; r < 8; ++r) partial[wave][r] = acc[r];
  } else if (lane == 16) {
#pragma unroll
    for (int r = 0; r < 8; ++r) partial[wave][8 + r] = acc[r];
  }
  __syncthreads();

  // sequential LIF scan + exact JAX RNG on one lane
  if (tid == 0) {
    const float decay = decay_p[0];
    const float vth   = vth_p[0];
    float u = 0.0f, o = 0.0f;
#pragma unroll
    for (int t = 0; t < T_STEPS; ++t) {
      float s = 0.0f;
#pragma unroll
      for (int w = 0; w < 8; ++w) s += partial[w][t];
      const float inp = s * (1.0f / (float)HW);       // mean over 56*56
      u = decay * u + inp - o * vth;
      uint32_t k0, k1; jax_step_key(t, &k0, &k1);
      const float r = jax_uniform4096(k0, k1, (uint32_t)bc);
      const float p = 1.0f - __expf(-0.1f * (u - vth));   // BETA = 0.1
      o = (p > r) ? 1.0f : 0.0f;
      out[bc * T_STEPS + t] = o;
    }
  }
}

extern "C" void kernel_launch(void* const* d_in, const int* in_sizes, int n_in,
                              void* d_out, int out_size, void* d_ws, size_t ws_size,
                              hipStream_t stream) {
  (void)in_sizes; (void)n_in; (void)out_size; (void)d_ws; (void)ws_size;
  const float* x     = (const float*)d_in[0];
  const float* decay = (const float*)d_in[1];
  const float* vth   = (const float*)d_in[2];
  float* out = (float*)d_out;
  hipLaunchKernelGGL(snn_gap_lif_kernel, dim3(16 * 256), dim3(NTHREADS), 0, stream,
                     x, decay, vth, out);
}